// AutoNet_23063974379760
// MI455X (gfx1250) — compile-verified
//
#include <hip/hip_runtime.h>
#include <math.h>
#include <stdint.h>

// ---------------- problem constants ----------------
#define S_   2
#define T_   16
#define STEP_ 5
#define LAT_ 1000
#define HID_ 300
#define GATES_ 1200            // 4*HID
#define IMGF_ 61440            // 3*128*160 per camera
#define ROWS_ 192              // S*T*6
#define FCO_  10000            // 16*25*25
#define BN_SCALE_F 0.9999950000374997f   // 1/sqrt(1+1e-5)

typedef float v2f __attribute__((ext_vector_type(2)));
typedef float v8f __attribute__((ext_vector_type(8)));

static __device__ __forceinline__ v8f wmma4(v2f a, v2f b, v8f c) {
  // D = A(16x4) * B(4x16) + C, fp32 WMMA (wave32)
  return __builtin_amdgcn_wmma_f32_16x16x4_f32(false, a, false, b, (short)0, c, false, false);
}
static __device__ __forceinline__ float sigmoidf_(float x) {
  return 1.f / (1.f + expf(-x));
}

// async 16B global -> LDS copy (ASYNCcnt-tracked, no VGPR round trip).
// LDS operand is the low 32 bits of the generic pointer.
static __device__ __forceinline__ void async_cp16(const float* __restrict__ gsrc,
                                                  float* ldst) {
  const uint32_t l = (uint32_t)(uintptr_t)ldst;
  asm volatile("global_load_async_to_lds_b128 %0, %1, off"
               :: "v"(l), "v"(gsrc)
               : "memory");
}

// =====================================================================
// Kernel 1: backbone GEMM (mu half only).  part[ks][192][1000] partials.
// grid = (63 n-tiles, 4 k-slices), block = 416 (13 waves; waves 0..11
// are the 12 M-tiles).  Double-buffered LDS fed by async-to-LDS copies:
// per 64-wide K chunk: 208 rows * 16 b128 = 3328 transfers = exactly 8
// per thread, so "s_wait_asynccnt 8" after issuing chunk c+1 guarantees
// chunk c is resident (per-wave in-order completion).
// LDS row stride 68 floats: 16B-aligned rows AND conflict-free 16-lane
// column reads (68 mod 64 = 4 -> distinct banks).
// =====================================================================
#define ASTRIDE 68
#define A_FLOATS (ROWS_ * ASTRIDE)   // 13056
#define B_FLOATS (16 * ASTRIDE)      // 1088

__global__ __launch_bounds__(416) void k_backbone(const float* __restrict__ X,
                                                  const float* __restrict__ W,
                                                  float* __restrict__ part) {
  extern __shared__ float sm[];
  float* A0 = sm;
  float* B0 = A0 + A_FLOATS;
  float* A1 = B0 + B_FLOATS;
  float* B1 = A1 + A_FLOATS;

  const int nt = blockIdx.x;          // 0..62
  const int ks = blockIdx.y;          // 0..3
  const int tid = threadIdx.x;
  const int lane = tid & 31;
  const int wv = tid >> 5;            // 0..12 (0..11 compute M-tiles)
  const int m0 = wv * 16;
  const int ml = lane & 15;
  const int khalf = (lane >> 4) * 2;  // 0 or 2
  v8f acc = {0.f,0.f,0.f,0.f,0.f,0.f,0.f,0.f};

  const int kbeg = ks * 15360;        // 61440/4

  // issue one chunk's async copies: A rows 0..191, B rows 192..207
  auto issue = [&](int c, float* Ab, float* Bb) {
    const int k0 = kbeg + c * 64;
#pragma unroll
    for (int j = 0; j < 8; ++j) {
      const int i = tid + j * 416;    // 0..3327
      const int row = i >> 4, q = i & 15;
      if (row < ROWS_) {
        async_cp16(X + (size_t)row * IMGF_ + k0 + q * 4, Ab + row * ASTRIDE + q * 4);
      } else {
        const int r2 = row - ROWS_;   // 0..15 weight rows
        async_cp16(W + (size_t)(nt * 16 + r2) * IMGF_ + k0 + q * 4,
                   Bb + r2 * ASTRIDE + q * 4);
      }
    }
  };

  issue(0, A0, B0);
  for (int c = 0; c < 240; ++c) {
    float* Ab = (c & 1) ? A1 : A0;
    float* Bb = (c & 1) ? B1 : B0;
    if (c + 1 < 240) {
      issue(c + 1, (c & 1) ? A0 : A1, (c & 1) ? B0 : B1);
      asm volatile("s_wait_asynccnt 8" ::: "memory");   // chunk c resident
    } else {
      asm volatile("s_wait_asynccnt 0" ::: "memory");
    }
    __syncthreads();
    if (wv < 12) {
#pragma unroll
      for (int s = 0; s < 16; ++s) {
        const int kl = s * 4 + khalf;
        const float* pa = &Ab[(m0 + ml) * ASTRIDE + kl];
        const float* pb = &Bb[ml * ASTRIDE + kl];
        v2f a = {pa[0], pa[1]};
        v2f b = {pb[0], pb[1]};
        acc = wmma4(a, b, acc);
      }
    }
    __syncthreads();
  }
  const int n = nt * 16 + ml;
  if (wv < 12 && n < LAT_) {
    const int rbase = (lane >> 4) * 8;
#pragma unroll
    for (int vv = 0; vv < 8; ++vv) {
      const int row = m0 + vv + rbase;
      part[((size_t)ks * ROWS_ + row) * LAT_ + n] = acc[vv];
    }
  }
}

// reduce 4 K-slices + bias -> mu[192][1000]
__global__ void k_reduce_mu(const float* __restrict__ part,
                            const float* __restrict__ bias,
                            float* __restrict__ mu) {
  const int i = blockIdx.x * blockDim.x + threadIdx.x;
  if (i < ROWS_ * LAT_) {
    const int n = i % LAT_;
    mu[i] = part[i] + part[192000 + i] + part[384000 + i] + part[576000 + i] + bias[n];
  }
}

// =====================================================================
// Kernel 2: one LSTM chunk (4 windows in parallel, both scenes packed
// into one WMMA M: scene0 -> rows 0..5, scene1 -> rows 8..13).
// grid = 4 (window), block = 512 (16 waves), launched once per chunk.
// dynamic LDS: x[2][6][1000] | g[2][6][1200] | h1,c1,h2,c2 [2][6][300]
// =====================================================================
__global__ __launch_bounds__(512) void k_lstm_chunk(
    const float* __restrict__ mu,
    const float* __restrict__ Wih1, const float* __restrict__ Whh1,
    const float* __restrict__ bih1, const float* __restrict__ bhh1,
    const float* __restrict__ Wih2, const float* __restrict__ Whh2,
    const float* __restrict__ bih2, const float* __restrict__ bhh2,
    const float* __restrict__ c_in, float* __restrict__ c_out,
    float* __restrict__ lstm_out, int chunk) {
  extern __shared__ float sm[];
  float* x_s = sm;             // 12000
  float* g_s = x_s + 12000;    // 14400
  float* h1s = g_s + 14400;    // 3600 each
  float* c1s = h1s + 3600;
  float* h2s = c1s + 3600;
  float* c2s = h2s + 3600;

  const int w = blockIdx.x;            // window-in-chunk 0..3
  const int t = chunk * 4 + w;         // frame index
  const int tid = threadIdx.x;
  const int lane = tid & 31;
  const int wv = tid >> 5;             // 0..15
  const int ml = lane & 15;
  const int khalf = (lane >> 4) * 2;
  const int scn = ml >> 3, rr = ml & 7;
  const bool rowValid = (rr < 6);
  const int rbase = (lane >> 4) * 8;

  // init state: h=0 every window (reference "bug"); c from chain
  for (int i = tid; i < 3600; i += 512) {
    float a = 0.f, b = 0.f;
    if (chunk > 0) {
      const int s = i / 1800, r = i % 1800;
      a = c_in[(s * 2 + 0) * 1800 + r];
      b = c_in[(s * 2 + 1) * 1800 + r];
    }
    c1s[i] = a; c2s[i] = b; h1s[i] = 0.f; h2s[i] = 0.f;
  }
  __syncthreads();

  for (int p = 0; p < STEP_; ++p) {
    const int f = t - 4 + p;           // source frame (may be padding)
    for (int i = tid; i < 12000; i += 512) {
      const int s = i / 6000, rem = i % 6000;
      x_s[i] = (f >= 0) ? mu[((size_t)(s * 16 + f) * 6) * LAT_ + rem] : 0.f;
    }
    __syncthreads();

    // ---- layer 1 gates: g1 = [x | h1] (K=1300) x [Wih1|Whh1]^T ----
    {
      v8f acc[5];
#pragma unroll
      for (int i = 0; i < 5; ++i) acc[i] = (v8f){0.f,0.f,0.f,0.f,0.f,0.f,0.f,0.f};
      for (int kk = 0; kk < 1300; kk += 4) {
        const int kl = kk + khalf;
        v2f a;
        if (!rowValid) { a = (v2f){0.f, 0.f}; }
        else if (kl < LAT_) {
          const float* p0 = &x_s[(scn * 6 + rr) * LAT_ + kl];
          a = (v2f){p0[0], p0[1]};
        } else {
          const float* p0 = &h1s[(scn * 6 + rr) * HID_ + (kl - LAT_)];
          a = (v2f){p0[0], p0[1]};
        }
#pragma unroll
        for (int i = 0; i < 5; ++i) {
          const int nt = wv + i * 16;
          if (nt < 75) {
            const int n = nt * 16 + ml;
            const float* pw = (kl < LAT_) ? &Wih1[(size_t)n * LAT_ + kl]
                                          : &Whh1[(size_t)n * HID_ + (kl - LAT_)];
            v2f b = (v2f){pw[0], pw[1]};
            acc[i] = wmma4(a, b, acc[i]);
          }
        }
      }
#pragma unroll
      for (int i = 0; i < 5; ++i) {
        const int nt = wv + i * 16;
        if (nt < 75) {
          const int n = nt * 16 + ml;
          const float bb = bih1[n] + bhh1[n];
#pragma unroll
          for (int vv = 0; vv < 8; ++vv) {
            const int row = vv + rbase;
            const int s = row >> 3, r = row & 7;
            if (r < 6) g_s[(s * 6 + r) * GATES_ + n] = acc[i][vv] + bb;
          }
        }
      }
    }
    __syncthreads();
    // layer 1 elementwise (torch gate order i,f,g,o)
    for (int i = tid; i < 3600; i += 512) {
      const int s = i / 1800, rem = i % 1800, r = rem / 300, j = rem % 300;
      const float* g = &g_s[(s * 6 + r) * GATES_];
      const float ig = sigmoidf_(g[j]);
      const float fg = sigmoidf_(g[300 + j]);
      const float gg = tanhf(g[600 + j]);
      const float og = sigmoidf_(g[900 + j]);
      const float cn = fg * c1s[i] + ig * gg;
      c1s[i] = cn;
      h1s[i] = og * tanhf(cn);
    }
    __syncthreads();

    // ---- layer 2 gates: g2 = [h1 | h2] (K=600) x [Wih2|Whh2]^T ----
    {
      v8f acc[5];
#pragma unroll
      for (int i = 0; i < 5; ++i) acc[i] = (v8f){0.f,0.f,0.f,0.f,0.f,0.f,0.f,0.f};
      for (int kk = 0; kk < 600; kk += 4) {
        const int kl = kk + khalf;
        v2f a;
        if (!rowValid) { a = (v2f){0.f, 0.f}; }
        else if (kl < HID_) {
          const float* p0 = &h1s[(scn * 6 + rr) * HID_ + kl];
          a = (v2f){p0[0], p0[1]};
        } else {
          const float* p0 = &h2s[(scn * 6 + rr) * HID_ + (kl - HID_)];
          a = (v2f){p0[0], p0[1]};
        }
#pragma unroll
        for (int i = 0; i < 5; ++i) {
          const int nt = wv + i * 16;
          if (nt < 75) {
            const int n = nt * 16 + ml;
            const float* pw = (kl < HID_) ? &Wih2[(size_t)n * HID_ + kl]
                                          : &Whh2[(size_t)n * HID_ + (kl - HID_)];
            v2f b = (v2f){pw[0], pw[1]};
            acc[i] = wmma4(a, b, acc[i]);
          }
        }
      }
#pragma unroll
      for (int i = 0; i < 5; ++i) {
        const int nt = wv + i * 16;
        if (nt < 75) {
          const int n = nt * 16 + ml;
          const float bb = bih2[n] + bhh2[n];
#pragma unroll
          for (int vv = 0; vv < 8; ++vv) {
            const int row = vv + rbase;
            const int s = row >> 3, r = row & 7;
            if (r < 6) g_s[(s * 6 + r) * GATES_ + n] = acc[i][vv] + bb;
          }
        }
      }
    }
    __syncthreads();
    // layer 2 elementwise
    for (int i = tid; i < 3600; i += 512) {
      const int s = i / 1800, rem = i % 1800, r = rem / 300, j = rem % 300;
      const float* g = &g_s[(s * 6 + r) * GATES_];
      const float ig = sigmoidf_(g[j]);
      const float fg = sigmoidf_(g[300 + j]);
      const float gg = tanhf(g[600 + j]);
      const float og = sigmoidf_(g[900 + j]);
      const float cn = fg * c2s[i] + ig * gg;
      c2s[i] = cn;
      h2s[i] = og * tanhf(cn);
    }
    __syncthreads();
  }

  // per-frame output = final h2 of this window
  for (int i = tid; i < 3600; i += 512) {
    const int s = i / 1800, rem = i % 1800;
    lstm_out[((size_t)(s * 16 + t)) * 1800 + rem] = h2s[i];
  }
  // chunk carry = last window's final cell state
  if (w == 3) {
    for (int i = tid; i < 3600; i += 512) {
      const int s = i / 1800, rem = i % 1800;
      c_out[(s * 2 + 0) * 1800 + rem] = c1s[i];
      c_out[(s * 2 + 1) * 1800 + rem] = c2s[i];
    }
  }
}

// =====================================================================
// Kernel 3: fc2 (no bias) + eval-BN + ReLU.  [32,1800] x [10000,1800]^T
// grid = 625 n-tiles, block = 64 (2 waves = 2 M-tiles of 16)
// =====================================================================
__global__ __launch_bounds__(64) void k_fc2(const float* __restrict__ A,
                                            const float* __restrict__ W,
                                            const float* __restrict__ g,
                                            const float* __restrict__ b,
                                            float* __restrict__ out) {
  const int nt = blockIdx.x;      // 0..624
  __shared__ float As[32 * 61];
  __shared__ float Bs[16 * 61];
  const int tid = threadIdx.x;
  const int lane = tid & 31;
  const int wv = tid >> 5;        // 0..1
  const int m0 = wv * 16;
  const int ml = lane & 15;
  const int khalf = (lane >> 4) * 2;
  v8f acc = {0.f,0.f,0.f,0.f,0.f,0.f,0.f,0.f};

  for (int c = 0; c < 30; ++c) {  // 30 chunks of K=60 (1800 total)
    const int k0 = c * 60;
    for (int i = tid; i < 1920; i += 64) {
      const int row = i / 60, kk = i % 60;
      As[row * 61 + kk] = A[(size_t)row * 1800 + k0 + kk];
    }
    for (int i = tid; i < 960; i += 64) {
      const int row = i / 60, kk = i % 60;
      Bs[row * 61 + kk] = W[(size_t)(nt * 16 + row) * 1800 + k0 + kk];
      if (c + 1 < 30)  // stream hint for next weight chunk
        __builtin_prefetch(W + (size_t)(nt * 16 + row) * 1800 + k0 + 60 + kk, 0, 1);
    }
    __syncthreads();
#pragma unroll
    for (int s = 0; s < 15; ++s) {
      const int kl = s * 4 + khalf;
      const float* pa = &As[(m0 + ml) * 61 + kl];
      const float* pb = &Bs[ml * 61 + kl];
      v2f a = {pa[0], pa[1]};
      v2f bb = {pb[0], pb[1]};
      acc = wmma4(a, bb, acc);
    }
    __syncthreads();
  }
  const int n = nt * 16 + ml;
  const float scale = g[n] * BN_SCALE_F;
  const float shift = b[n];
  const int rbase = (lane >> 4) * 8;
#pragma unroll
  for (int vv = 0; vv < 8; ++vv) {
    const int row = m0 + vv + rbase;
    const float v = acc[vv] * scale + shift;
    out[(size_t)row * FCO_ + n] = v > 0.f ? v : 0.f;
  }
}

// =====================================================================
// Kernel 4: ConvTranspose2d(k=3,s=2,p=1,op=1) + BN/ReLU or bias/sigmoid
// weight layout [Cin, Cout, 3, 3] (torch transposed-conv layout)
// =====================================================================
__global__ void k_deconv(const float* __restrict__ in, const float* __restrict__ wgt,
                         const float* __restrict__ bnG, const float* __restrict__ bnB,
                         const float* __restrict__ cbias, float* __restrict__ out,
                         int N, int Cin, int Cout, int Hin, int Win, int mode) {
  const int Hout = 2 * Hin, Wout = 2 * Win;
  const int total = N * Cout * Hout * Wout;
  const int idx = blockIdx.x * blockDim.x + threadIdx.x;
  if (idx >= total) return;
  const int ox = idx % Wout;
  int tmp = idx / Wout;
  const int oy = tmp % Hout; tmp /= Hout;
  const int co = tmp % Cout;
  const int n = tmp / Cout;
  float sum = 0.f;
  for (int ky = 0; ky < 3; ++ky) {
    const int ty = oy + 1 - ky;
    if (ty < 0 || (ty & 1)) continue;
    const int iy = ty >> 1;
    if (iy >= Hin) continue;
    for (int kx = 0; kx < 3; ++kx) {
      const int tx = ox + 1 - kx;
      if (tx < 0 || (tx & 1)) continue;
      const int ix = tx >> 1;
      if (ix >= Win) continue;
      for (int ci = 0; ci < Cin; ++ci) {
        sum += in[(((size_t)n * Cin + ci) * Hin + iy) * Win + ix] *
               wgt[((ci * Cout + co) * 3 + ky) * 3 + kx];
      }
    }
  }
  if (mode == 0) {
    const float v = sum * (bnG[co] * BN_SCALE_F) + bnB[co];
    out[idx] = v > 0.f ? v : 0.f;
  } else {
    const float v = sum + cbias[co];
    out[idx] = 1.f / (1.f + expf(-v));
  }
}

// =====================================================================
extern "C" void kernel_launch(void* const* d_in, const int* in_sizes, int n_in,
                              void* d_out, int out_size, void* d_ws, size_t ws_size,
                              hipStream_t stream) {
  const float* x          = (const float*)d_in[0];
  const float* backbone_w = (const float*)d_in[1];
  const float* backbone_b = (const float*)d_in[2];
  const float* Wih1 = (const float*)d_in[3];
  const float* Whh1 = (const float*)d_in[4];
  const float* bih1 = (const float*)d_in[5];
  const float* bhh1 = (const float*)d_in[6];
  const float* Wih2 = (const float*)d_in[7];
  const float* Whh2 = (const float*)d_in[8];
  const float* bih2 = (const float*)d_in[9];
  const float* bhh2 = (const float*)d_in[10];
  const float* fc2_w   = (const float*)d_in[11];
  const float* bn_fc_g = (const float*)d_in[12];
  const float* bn_fc_b = (const float*)d_in[13];
  const float* d0_w  = (const float*)d_in[14];
  const float* bn0_g = (const float*)d_in[15];
  const float* bn0_b = (const float*)d_in[16];
  const float* d1_w  = (const float*)d_in[17];
  const float* bn1_g = (const float*)d_in[18];
  const float* bn1_b = (const float*)d_in[19];
  const float* d2_w  = (const float*)d_in[20];
  const float* d2_b  = (const float*)d_in[21];
  float* outp = (float*)d_out;

  float* ws = (float*)d_ws;
  float* mu_part  = ws;                    // 4*192*1000 = 768000
  float* mu       = mu_part + 768000;      // 192000
  float* lstm_out = mu + 192000;           // 57600
  float* cbuf0    = lstm_out + 57600;      // 7200
  float* cbuf1    = cbuf0 + 7200;          // 7200
  float* fc_out   = cbuf1 + 7200;          // 320000
  float* dc0      = fc_out + 320000;       // 32*8*50*50   = 640000
  float* dc1      = dc0 + 640000;          // 32*4*100*100 = 1280000

  // 1) backbone GEMM (split-K=4, async double-buffered LDS) + reduce
  const size_t bb_lds = 2 * (A_FLOATS + B_FLOATS) * sizeof(float);  // ~113 KB
  k_backbone<<<dim3(63, 4), 416, bb_lds, stream>>>(x, backbone_w, mu_part);
  k_reduce_mu<<<(ROWS_ * LAT_ + 255) / 256, 256, 0, stream>>>(mu_part, backbone_b, mu);

  // 2) LSTM: 4 sequential chunks, cell state chained via double buffer
  const size_t lstm_lds = (12000 + 14400 + 4 * 3600) * sizeof(float);  // 163.2 KB
  for (int chunk = 0; chunk < 4; ++chunk) {
    float* c_in  = (chunk & 1) ? cbuf1 : cbuf0;
    float* c_out = (chunk & 1) ? cbuf0 : cbuf1;
    k_lstm_chunk<<<4, 512, lstm_lds, stream>>>(mu, Wih1, Whh1, bih1, bhh1,
                                               Wih2, Whh2, bih2, bhh2,
                                               c_in, c_out, lstm_out, chunk);
  }

  // 3) fc2 + BN + ReLU
  k_fc2<<<625, 64, 0, stream>>>(lstm_out, fc2_w, bn_fc_g, bn_fc_b, fc_out);

  // 4) deconv stack
  k_deconv<<<(32 * 8 * 50 * 50 + 255) / 256, 256, 0, stream>>>(
      fc_out, d0_w, bn0_g, bn0_b, nullptr, dc0, 32, 16, 8, 25, 25, 0);
  k_deconv<<<(32 * 4 * 100 * 100 + 255) / 256, 256, 0, stream>>>(
      dc0, d1_w, bn1_g, bn1_b, nullptr, dc1, 32, 8, 4, 50, 50, 0);
  k_deconv<<<(32 * 1 * 200 * 200 + 255) / 256, 256, 0, stream>>>(
      dc1, d2_w, nullptr, nullptr, d2_b, outp, 32, 4, 1, 100, 100, 1);
}